// VGAE_1778116461256
// MI455X (gfx1250) — compile-verified
//
#include <hip/hip_runtime.h>

typedef __attribute__((ext_vector_type(2))) float v2f;
typedef __attribute__((ext_vector_type(8))) float v8f;

#define N_NODES 50000

// ---------------- degree / symmetric normalization ----------------
__global__ void k_init_deg(float* deg, int n) {
  int i = blockIdx.x * blockDim.x + threadIdx.x;
  if (i < n) deg[i] = 1.0f;  // self-loop contribution
}

__global__ void k_count_deg(const long long* __restrict__ dst,
                            float* __restrict__ deg, int e) {
  int i = blockIdx.x * blockDim.x + threadIdx.x;
  if (i < e) atomicAdd(&deg[(int)dst[i]], 1.0f);
}

__global__ void k_dinv(float* deg, int n) {
  int i = blockIdx.x * blockDim.x + threadIdx.x;
  if (i < n) deg[i] = rsqrtf(deg[i]);  // deg >= 1 always (self loops)
}

// ---------------- fp32 WMMA GEMM: C[M,N] = A[M,K] @ B[K,N] ----------------
// K == 128 for both layers. One 16-row M-tile per block; 128 threads = 4
// wave32s; waves split the N-tiles. V_WMMA_F32_16X16X4_F32 keeps full fp32
// precision (workload is bandwidth-bound, low precision buys nothing).
// A tile is staged global->LDS with GLOBAL_LOAD_ASYNC_TO_LDS_B128 (ASYNCcnt).
#define LDS_STRIDE 132  // +4 dwords: 16B-aligned rows AND bank-conflict-free cols

__global__ __launch_bounds__(128) void k_gemm_wmma(const float* __restrict__ A,
                                                   const float* __restrict__ B,
                                                   float* __restrict__ C,
                                                   int M, int K, int N) {
  __shared__ float As[16 * LDS_STRIDE];
  const int tid = threadIdx.x;
  const int m0  = blockIdx.x * 16;

  // async staging of the 16 x K A-tile: 16-byte chunks, no VGPR round trip
  const unsigned ldsBase = (unsigned)(uintptr_t)As;  // low 32b = LDS byte offset
  const int chunksPerRow = K >> 2;                   // float4 chunks (32 for K=128)
  for (int i = tid; i < 16 * chunksPerRow; i += 128) {
    int r  = i / chunksPerRow;
    int c4 = i - r * chunksPerRow;
    int gr = m0 + r;
    unsigned lo = ldsBase + (unsigned)((r * LDS_STRIDE + c4 * 4) * 4);
    if (gr < M) {
      const float* gp = A + (size_t)gr * K + c4 * 4;
      asm volatile("global_load_async_to_lds_b128 %0, %1, off"
                   :: "v"(lo), "v"(gp) : "memory");
    } else {
      *(float4*)(As + r * LDS_STRIDE + c4 * 4) = make_float4(0.f, 0.f, 0.f, 0.f);
    }
  }
  asm volatile("s_wait_asynccnt 0x0" ::: "memory");
  __syncthreads();

  const int lane = tid & 31;
  const int wave = tid >> 5;
  const int lm   = lane & 15;  // A row / B,C column within tile
  const int g    = lane >> 4;  // half-wave selects K pair (ISA 16x4 layout)

  const int ntiles = N >> 4;
  for (int nt = wave; nt < ntiles; nt += 4) {
    const int n0 = nt * 16;
    v8f acc = {};
    for (int kk = 0; kk < K; kk += 4) {
      const int kb = kk + 2 * g;
      v2f a, b;
      // A 16x4 f32 layout: VGPR j, half-wave g -> K = 2g + j, M = lane&15
      a.x = As[lm * LDS_STRIDE + kb + 0];
      a.y = As[lm * LDS_STRIDE + kb + 1];
      // B 4x16 f32 layout: VGPR j, half-wave g -> K = 2g + j, N = lane&15
      b.x = B[(size_t)(kb + 0) * N + n0 + lm];
      b.y = B[(size_t)(kb + 1) * N + n0 + lm];
      acc = __builtin_amdgcn_wmma_f32_16x16x4_f32(false, a, false, b,
                                                  (short)0, acc, false, false);
    }
    // C/D 16x16 f32 layout: VGPR r -> row r + 8*g, col = lane&15
#pragma unroll
    for (int r = 0; r < 8; ++r) {
      int row = m0 + r + 8 * g;
      if (row < M) C[(size_t)row * N + n0 + lm] = acc[r];
    }
  }
}

// ---------------- GCN aggregation ----------------
// out[i][:] = dinv[i]^2 * h[i][:] + bias[:]  (self-loop term + bias), float4-wide
template <int F>
__global__ void k_agg_self(const float4* __restrict__ h4,
                           const float* __restrict__ dinv,
                           const float4* __restrict__ bias4,
                           float4* __restrict__ out4, long long n4) {
  long long i = (long long)blockIdx.x * blockDim.x + threadIdx.x;
  if (i >= n4) return;
  constexpr int F4 = F / 4;
  int node = (int)(i / F4);
  int f4   = (int)(i - (long long)node * F4);
  float di = dinv[node];
  float s  = di * di;
  float4 hv = h4[i];
  float4 bv = bias4[f4];
  out4[i] = make_float4(s * hv.x + bv.x, s * hv.y + bv.y,
                        s * hv.z + bv.z, s * hv.w + bv.w);
}

// one wave per edge: out[dst] += dinv[src]*dinv[dst] * h[src]
// gather is one b128/b64 per lane (full cachelines); scatter is f32 atomics
template <int F>
__global__ void k_agg_edges(const float* __restrict__ h,
                            const float* __restrict__ dinv,
                            const long long* __restrict__ src,
                            const long long* __restrict__ dst,
                            float* __restrict__ out, int e) {
  constexpr int VEC = F / 32;  // 4 for F=128, 2 for F=64 (wave32)
  int w    = (int)(((long long)blockIdx.x * blockDim.x + threadIdx.x) >> 5);
  int lane = threadIdx.x & 31;
  if (w >= e) return;
  int s = (int)src[w];
  const float* hs = h + (size_t)s * F + lane * VEC;
  __builtin_prefetch(hs, 0, 3);  // global_prefetch_b8: overlap the dinv gathers
  int d = (int)dst[w];
  float nrm = dinv[s] * dinv[d];
  float* od = out + (size_t)d * F + lane * VEC;
  if constexpr (VEC == 4) {
    float4 hv = *(const float4*)hs;
    atomicAdd(od + 0, nrm * hv.x);
    atomicAdd(od + 1, nrm * hv.y);
    atomicAdd(od + 2, nrm * hv.z);
    atomicAdd(od + 3, nrm * hv.w);
  } else {
    float2 hv = *(const float2*)hs;
    atomicAdd(od + 0, nrm * hv.x);
    atomicAdd(od + 1, nrm * hv.y);
  }
}

__global__ void k_relu4(float4* x, long long n4) {
  long long i = (long long)blockIdx.x * blockDim.x + threadIdx.x;
  if (i >= n4) return;
  float4 v = x[i];
  x[i] = make_float4(fmaxf(v.x, 0.f), fmaxf(v.y, 0.f),
                     fmaxf(v.z, 0.f), fmaxf(v.w, 0.f));
}

// ---------------- decoder: per-edge dot product over F=64 ----------------
__global__ void k_decode64(const float* __restrict__ z,
                           const long long* __restrict__ ia,
                           const long long* __restrict__ ib,
                           float* __restrict__ out, int e) {
  int w    = (int)(((long long)blockIdx.x * blockDim.x + threadIdx.x) >> 5);
  int lane = threadIdx.x & 31;
  if (w >= e) return;
  const float2* za = (const float2*)(z + (size_t)ia[w] * 64) + lane;
  const float2* zb = (const float2*)(z + (size_t)ib[w] * 64) + lane;
  float2 a = *za;
  float2 b = *zb;
  float acc = a.x * b.x + a.y * b.y;
#pragma unroll
  for (int off = 16; off > 0; off >>= 1) acc += __shfl_xor(acc, off, 32);
  if (lane == 0) out[w] = acc;
}

// ---------------- launch ----------------
extern "C" void kernel_launch(void* const* d_in, const int* in_sizes, int n_in,
                              void* d_out, int out_size, void* d_ws, size_t ws_size,
                              hipStream_t stream) {
  const float*     x   = (const float*)d_in[0];
  const long long* ei  = (const long long*)d_in[1];  // int64 edge_index [2,E]
  const long long* eli = (const long long*)d_in[2];  // int64 edge_label_index [2,EL]
  const float*     W1  = (const float*)d_in[3];
  const float*     b1  = (const float*)d_in[4];
  const float*     W2  = (const float*)d_in[5];
  const float*     b2  = (const float*)d_in[6];
  float* out = (float*)d_out;

  const int N  = N_NODES;
  const int Fin = 128, Fh = 128, Fo = 64;
  const int E  = in_sizes[1] / 2;   // 800000
  const int EL = in_sizes[2] / 2;   // 200000
  const long long* src = ei;
  const long long* dst = ei + E;
  const long long* la  = eli;
  const long long* lb  = eli + EL;

  auto alignup = [](size_t v) { return (v + 255) & ~(size_t)255; };
  char* p = (char*)d_ws;
  float* dinv = (float*)p; p += alignup((size_t)N * 4);
  float* h1   = (float*)p; p += alignup((size_t)N * Fh * 4);  // x @ W1
  float* g1   = (float*)p; p += alignup((size_t)N * Fh * 4);  // relu(agg1 + b1)
  float* h2   = (float*)p; p += alignup((size_t)N * Fo * 4);  // g1 @ W2
  float* g2   = (float*)p;                                    // agg2 + b2 (= z)

  // degrees -> dinv (in place)
  k_init_deg <<<(N + 255) / 256, 256, 0, stream>>>(dinv, N);
  k_count_deg<<<(E + 255) / 256, 256, 0, stream>>>(dst, dinv, E);
  k_dinv     <<<(N + 255) / 256, 256, 0, stream>>>(dinv, N);

  // layer 1: linear (WMMA) -> self+bias -> edge scatter -> relu
  k_gemm_wmma<<<(N + 15) / 16, 128, 0, stream>>>(x, W1, h1, N, Fin, Fh);
  {
    long long n4 = (long long)N * Fh / 4;
    k_agg_self<Fh><<<(unsigned)((n4 + 255) / 256), 256, 0, stream>>>(
        (const float4*)h1, dinv, (const float4*)b1, (float4*)g1, n4);
    k_agg_edges<Fh><<<(E + 7) / 8, 256, 0, stream>>>(h1, dinv, src, dst, g1, E);
    k_relu4<<<(unsigned)((n4 + 255) / 256), 256, 0, stream>>>((float4*)g1, n4);
  }

  // layer 2
  k_gemm_wmma<<<(N + 15) / 16, 128, 0, stream>>>(g1, W2, h2, N, Fh, Fo);
  {
    long long n4 = (long long)N * Fo / 4;
    k_agg_self<Fo><<<(unsigned)((n4 + 255) / 256), 256, 0, stream>>>(
        (const float4*)h2, dinv, (const float4*)b2, (float4*)g2, n4);
    k_agg_edges<Fo><<<(E + 7) / 8, 256, 0, stream>>>(h2, dinv, src, dst, g2, E);
  }

  // per-edge dot-product decoder
  k_decode64<<<(EL + 7) / 8, 256, 0, stream>>>(g2, la, lb, out, EL);
}